// LB_CNN_MonteCarlo_34840774705392
// MI455X (gfx1250) — compile-verified
//
#include <hip/hip_runtime.h>
#include <hip/hip_bf16.h>

// Problem constants (from reference)
#define S_ 32
#define B_ 128
#define M_ 128
#define F_ 512
#define C_ 100
#define N2_ 256   // two sign-tables of M_ columns concatenated

typedef __attribute__((ext_vector_type(2))) float v2f;
typedef __attribute__((ext_vector_type(8))) float v8f;

// ---------------------------------------------------------------------------
// Kernel 1: eta[b,m] = sigmoid(alpha_0[m] + sum_f x[b,f]*alpha[m,f])
// One wave32 per 16x16 tile, V_WMMA_F32_16X16X4_F32, K-step 4.
// ---------------------------------------------------------------------------
__global__ __launch_bounds__(32) void eta_kernel(const float* __restrict__ x,
                                                 const float* __restrict__ alpha,
                                                 const float* __restrict__ alpha_0,
                                                 float* __restrict__ eta) {
    const int lane = threadIdx.x;
    const int r    = lane & 15;
    const int half = lane >> 4;
    const int m0   = blockIdx.x * 16;
    const int b0   = blockIdx.y * 16;

    const float* xrow = x     + (size_t)(b0 + r) * F_;
    const float* arow = alpha + (size_t)(m0 + r) * F_;

    v8f acc = {};
#pragma unroll 4
    for (int f0 = 0; f0 < F_; f0 += 4) {
        v2f a = *(const v2f*)(xrow + f0 + half * 2);
        v2f b = *(const v2f*)(arow + f0 + half * 2);
        acc = __builtin_amdgcn_wmma_f32_16x16x4_f32(false, a, false, b,
                                                    (short)0, acc, false, false);
    }

    const int   n  = lane & 15;
    const float a0 = alpha_0[m0 + n];
#pragma unroll
    for (int i = 0; i < 8; ++i) {
        const float v = acc[i] + a0;
        eta[(size_t)(b0 + i + 8 * half) * M_ + (m0 + n)] = 1.0f / (1.0f + __expf(-v));
    }
}

// ---------------------------------------------------------------------------
// Kernel 2: base[(s*B+b), c] = beta_0[c] + sum_m Z[s,b,m]*beta[c,m]
// Rows = 4096, Cols = 100 (7x16 tiles with guards), K = 128.
// ---------------------------------------------------------------------------
__global__ __launch_bounds__(32) void base_kernel(const float* __restrict__ Z,
                                                  const float* __restrict__ beta,
                                                  const float* __restrict__ beta_0,
                                                  float* __restrict__ base) {
    const int lane = threadIdx.x;
    const int r    = lane & 15;
    const int half = lane >> 4;
    const int c0   = blockIdx.x * 16;
    const int r0   = blockIdx.y * 16;

    const float* zrow = Z + (size_t)(r0 + r) * M_;
    const int    c    = c0 + r;
    const bool   cv   = (c < C_);
    const float* brow = beta + (size_t)(cv ? c : 0) * M_;

    v8f acc = {};
#pragma unroll 4
    for (int k0 = 0; k0 < M_; k0 += 4) {
        v2f a = *(const v2f*)(zrow + k0 + half * 2);
        v2f b = {0.0f, 0.0f};
        if (cv) b = *(const v2f*)(brow + k0 + half * 2);
        acc = __builtin_amdgcn_wmma_f32_16x16x4_f32(false, a, false, b,
                                                    (short)0, acc, false, false);
    }

    const int n = lane & 15;
    if (c0 + n < C_) {
        const float b0v = beta_0[c0 + n];
#pragma unroll
        for (int i = 0; i < 8; ++i)
            base[(size_t)(r0 + i + 8 * half) * C_ + (c0 + n)] = acc[i] + b0v;
    }
}

// ---------------------------------------------------------------------------
// Kernel 3: in-place E = exp(base); rdenom[row] = 1 / sum_c E[row,c]
// (base is fully rewritten by base_kernel each call -> deterministic in-place)
// ---------------------------------------------------------------------------
__global__ __launch_bounds__(128) void exp_kernel(float* __restrict__ eb,
                                                  float* __restrict__ rdenom) {
    __shared__ float s_red[128];
    const int row = blockIdx.x;
    const int tid = threadIdx.x;

    float e = 0.0f;
    if (tid < C_) {
        e = __expf(eb[(size_t)row * C_ + tid]);
        eb[(size_t)row * C_ + tid] = e;
    }
    s_red[tid] = e;
    __syncthreads();
#pragma unroll
    for (int off = 64; off >= 1; off >>= 1) {
        if (tid < off) s_red[tid] += s_red[tid + off];
        __syncthreads();
    }
    if (tid == 0) rdenom[row] = 1.0f / s_red[0];
}

// ---------------------------------------------------------------------------
// Kernel 4: tabT[j, c] (row-major [256 x 100])
//   j <  128: exp(-2*beta[c, j])       (used when Z=+1, d=-2)
//   j >= 128: exp(+2*beta[c, j-128])   (used when Z=-1, d=+2)
// ---------------------------------------------------------------------------
__global__ __launch_bounds__(256) void tab_kernel(const float* __restrict__ beta,
                                                  float* __restrict__ tabT) {
    const int idx = blockIdx.x * 256 + threadIdx.x;   // < 25600
    const int j   = idx / C_;
    const int c   = idx - j * C_;
    const int m   = j & (M_ - 1);
    const float sgn = (j < M_) ? -2.0f : 2.0f;
    tabT[idx] = __expf(sgn * beta[(size_t)c * M_ + m]);
}

// ---------------------------------------------------------------------------
// Kernel 5: den[row, j] = sum_c E[row, c] * tabT[j, c]
// 4096 x 256 x 100 f32 WMMA GEMM. 16x32 tile per wave: A operand shared by
// two accumulators -> 3 loads : 2 WMMAs per K-step instead of 2:1.
// K = 100 = 25 * 4 exactly.
// ---------------------------------------------------------------------------
__global__ __launch_bounds__(32) void den_kernel(const float* __restrict__ E,
                                                 const float* __restrict__ tabT,
                                                 float* __restrict__ den) {
    const int lane = threadIdx.x;
    const int r    = lane & 15;
    const int half = lane >> 4;
    const int n0   = blockIdx.x * 32;   // 0..224 (8 tiles of 32)
    const int r0   = blockIdx.y * 16;   // 0..4080

    const float* arow  = E    + (size_t)(r0 + r) * C_;
    const float* brow0 = tabT + (size_t)(n0 + r)      * C_;
    const float* brow1 = tabT + (size_t)(n0 + 16 + r) * C_;

    v8f acc0 = {};
    v8f acc1 = {};
#pragma unroll 5
    for (int k0 = 0; k0 < C_; k0 += 4) {
        const int off = k0 + half * 2;
        v2f a  = *(const v2f*)(arow  + off);
        v2f b0 = *(const v2f*)(brow0 + off);
        v2f b1 = *(const v2f*)(brow1 + off);
        acc0 = __builtin_amdgcn_wmma_f32_16x16x4_f32(false, a, false, b0,
                                                     (short)0, acc0, false, false);
        acc1 = __builtin_amdgcn_wmma_f32_16x16x4_f32(false, a, false, b1,
                                                     (short)0, acc1, false, false);
    }

    const int n = lane & 15;
#pragma unroll
    for (int i = 0; i < 8; ++i) {
        const size_t rowoff = (size_t)(r0 + i + 8 * half) * N2_;
        den[rowoff + n0 + n]      = acc0[i];
        den[rowoff + n0 + 16 + n] = acc1[i];
    }
}

// ---------------------------------------------------------------------------
// Kernel 6: elementwise finish. Block = b, thread = m, loop over s.
//   z = Z[s,b,m];  j = (z>0) ? m : 128+m
//   pi_flip = E[row,y]*tabT[j,y] / den[row,j];  pi_base = E[row,y]*rdenom[row]
//   w = 0.5 + z*(eta-0.5);  acc += pi_base*w + pi_flip*(1-w)
// ---------------------------------------------------------------------------
__global__ __launch_bounds__(128) void final_kernel(const float* __restrict__ Z,
                                                    const int*   __restrict__ y,
                                                    const float* __restrict__ eta,
                                                    const float* __restrict__ E,
                                                    const float* __restrict__ rdenom,
                                                    const float* __restrict__ tabT,
                                                    const float* __restrict__ den,
                                                    float* __restrict__ out) {
    __shared__ float s_ty[N2_];   // tabT[:, y_b] gathered once per block
    const int tid = threadIdx.x;  // m
    const int b   = blockIdx.x;
    const int yb  = y[b];

    s_ty[tid]       = tabT[(size_t)tid * C_ + yb];
    s_ty[tid + 128] = tabT[(size_t)(tid + 128) * C_ + yb];
    const float eta_m = eta[(size_t)b * M_ + tid];
    float acc = 0.0f;
    __syncthreads();

    for (int s = 0; s < S_; ++s) {
        const int row = s * B_ + b;
        const float z  = Z[(size_t)row * M_ + tid];
        if (s + 1 < S_)
            __builtin_prefetch(&Z[(size_t)(row + B_) * M_ + tid], 0, 1);

        const int   j  = (z > 0.0f) ? tid : (tid + 128);
        const float dn = den[(size_t)row * N2_ + j];
        const float Ey = E[(size_t)row * C_ + yb];        // broadcast load

        const float pi_base = Ey * rdenom[row];
        const float pi_flip = Ey * s_ty[j] / dn;
        const float w       = 0.5f + z * (eta_m - 0.5f);
        acc += pi_base * w + pi_flip * (1.0f - w);
    }
    out[(size_t)b * M_ + tid] = acc;
}

// ---------------------------------------------------------------------------
// Launch. Inputs: x, y, Z, alpha_0, alpha, beta_0, beta.
// Workspace (floats): eta 16384 | base/E 409600 | rdenom 4096 | tabT 25600
//                     | den 1048576  -> ~6.0 MB
// ---------------------------------------------------------------------------
extern "C" void kernel_launch(void* const* d_in, const int* in_sizes, int n_in,
                              void* d_out, int out_size, void* d_ws, size_t ws_size,
                              hipStream_t stream) {
    (void)in_sizes; (void)n_in; (void)out_size; (void)ws_size;

    const float* x       = (const float*)d_in[0];
    const int*   y       = (const int*)  d_in[1];
    const float* Z       = (const float*)d_in[2];
    const float* alpha_0 = (const float*)d_in[3];
    const float* alpha   = (const float*)d_in[4];
    const float* beta_0  = (const float*)d_in[5];
    const float* beta    = (const float*)d_in[6];
    float*       out     = (float*)d_out;

    float* eta    = (float*)d_ws;
    float* eb     = eta    + (size_t)B_ * M_;          // base, then E in-place
    float* rdenom = eb     + (size_t)S_ * B_ * C_;
    float* tabT   = rdenom + (size_t)S_ * B_;
    float* den    = tabT   + (size_t)N2_ * C_;

    eta_kernel  <<<dim3(M_ / 16, B_ / 16),               32, 0, stream>>>(x, alpha, alpha_0, eta);
    base_kernel <<<dim3((C_ + 15) / 16, (S_ * B_) / 16), 32, 0, stream>>>(Z, beta, beta_0, eb);
    exp_kernel  <<<dim3(S_ * B_),                       128, 0, stream>>>(eb, rdenom);
    tab_kernel  <<<dim3((N2_ * C_) / 256),              256, 0, stream>>>(beta, tabT);
    den_kernel  <<<dim3(N2_ / 32, (S_ * B_) / 16),       32, 0, stream>>>(eb, tabT, den);
    final_kernel<<<dim3(B_),                            128, 0, stream>>>(Z, y, eta, eb, rdenom,
                                                                          tabT, den, out);
}